// LocallyConnected2d_34059090657757
// MI455X (gfx1250) — compile-verified
//
#include <hip/hip_runtime.h>

typedef __attribute__((ext_vector_type(2))) float v2f;
typedef __attribute__((ext_vector_type(8))) float v8f;
typedef __attribute__((ext_vector_type(4))) float f4;

#define Nb  32          // batch
#define Cc  32          // in channels
#define Hh  56
#define Ww  56
#define Oo  32          // out channels
#define CKK 288         // C*3*3
#define PP  3136        // H*W
#define TW  4           // w-positions per workgroup

// out[n,o,h,w] = sum_{c,ki,kj} wgt[o,h,w,c,ki,kj] * x[n,c,h+ki-1,w+kj-1] + bias[o,h,w]
// Per position: GEMM  D[32n x 32o] = P[32n x 288k] * W[288k x 32o], fp32 WMMA 16x16x4.
__global__ __launch_bounds__(128) void lc2d_wmma_f32(
    const float* __restrict__ x,     // [N][C][H][W]
    const float* __restrict__ wgt,   // [O][H][W][C][3][3]
    const float* __restrict__ bias,  // [1][O][H][W]
    float* __restrict__ out)         // [N][O][H][W]
{
    __shared__ float sP[CKK][33];        // im2col patches, [k][n], pad 33 -> conflict-free
    __shared__ float sW[CKK][33];        // weights, [k][o]
    __shared__ float sO[Nb][33][TW];     // staged output for coalesced float4 stores

    const int t     = threadIdx.x;       // 0..127
    const int lane  = t & 31;
    const int wave  = t >> 5;            // 0..3
    const int half  = lane >> 4;         // K-half select per WMMA layout
    const int m16   = lane & 15;
    const int mtile = wave >> 1;         // batch tile (rows)
    const int ntile = wave & 1;          // o tile (cols)
    const int mA    = m16 + 16 * mtile;  // batch row this lane feeds into A
    const int oC    = m16 + 16 * ntile;  // output channel column for B / C / D

    const int h  = blockIdx.y;
    const int w0 = blockIdx.x * TW;

    for (int tw = 0; tw < TW; ++tw) {
        const int w = w0 + tw;
        const int p = h * Ww + w;

        __syncthreads();  // previous iteration's readers done before restaging

        // ---- stage weight: sW[k][o] = wgt[(o*PP + p)*CKK + k]
        // wave handles 8 o's; lanes sweep k in 9 coalesced 128B chunks
        for (int oo = 0; oo < 8; ++oo) {
            const int o = wave * 8 + oo;
            const float* src = wgt + ((size_t)o * PP + p) * CKK;
#pragma unroll
            for (int s = 0; s < 9; ++s) {
                const int k = s * 32 + lane;
                sW[k][o] = src[k];
            }
        }

        // ---- stage patches: sP[k][n] = x[n][c][h+ki-1][w+kj-1]  (0 if OOB)
        for (int e = 0; e < 72; ++e) {
            const int idx = e * 128 + t;    // 9216 elements total
            const int n   = idx & 31;
            const int k   = idx >> 5;       // 0..287 = c*9 + ki*3 + kj
            const int c   = k / 9;
            const int r9  = k - c * 9;
            const int ki  = r9 / 3;
            const int kj  = r9 - ki * 3;
            const int hh  = h + ki - 1;
            const int ww  = w + kj - 1;
            float v = 0.0f;
            if ((unsigned)hh < (unsigned)Hh && (unsigned)ww < (unsigned)Ww)
                v = x[(((size_t)n * Cc + c) * Hh + hh) * Ww + ww];
            sP[k][n] = v;
        }

        // prefetch a slice of next position's weight while we compute (global_prefetch_b8)
        if (tw + 1 < TW) {
            const int o = wave * 8 + (lane >> 2);
            __builtin_prefetch(wgt + ((size_t)o * PP + (p + 1)) * CKK + (size_t)(lane & 3) * 72, 0, 1);
        }

        __syncthreads();

        // ---- GEMM 32x32x288: each wave computes one 16x16 tile with fp32 WMMA, K in steps of 4.
        // A layout (16x4 f32): VGPR v, lanes[0:15] K=k0+v, lanes[16:31] K=k0+v+2, M=lane%16.
        // B layout (4x16 f32): VGPR v, K=k0+v+2*half, N=lane%16.
        v8f acc = {};
#pragma unroll
        for (int k0 = 0; k0 < CKK; k0 += 4) {
            const int kA = k0 + 2 * half;
            v2f a, b;
            a[0] = sP[kA + 0][mA];
            a[1] = sP[kA + 1][mA];
            b[0] = sW[kA + 0][oC];
            b[1] = sW[kA + 1][oC];
            acc = __builtin_amdgcn_wmma_f32_16x16x4_f32(
                false, a, false, b, (short)0, acc, false, false);
        }

        // ---- bias + stage to LDS (C/D layout: VGPR r -> M = r + 8*half, N = lane%16)
        const float bv = bias[(size_t)oC * PP + p];
#pragma unroll
        for (int r = 0; r < 8; ++r) {
            const int n = r + 8 * half + 16 * mtile;
            sO[n][oC][tw] = acc[r] + bv;
        }
    }

    __syncthreads();

    // ---- coalesced output: float4 along w (w0 multiple of 4 -> 16B aligned)
    for (int e = 0; e < 8; ++e) {
        const int pair = e * 128 + t;     // 1024 (n,o) pairs
        const int n = pair >> 5;
        const int o = pair & 31;
        f4 v;
        v.x = sO[n][o][0];
        v.y = sO[n][o][1];
        v.z = sO[n][o][2];
        v.w = sO[n][o][3];
        *(f4*)(out + ((size_t)n * Oo + o) * PP + (size_t)h * Ww + w0) = v;
    }
}

extern "C" void kernel_launch(void* const* d_in, const int* in_sizes, int n_in,
                              void* d_out, int out_size, void* d_ws, size_t ws_size,
                              hipStream_t stream) {
    const float* x    = (const float*)d_in[0];  // 32*32*56*56
    const float* wgt  = (const float*)d_in[1];  // 32*56*56*32*3*3
    const float* bias = (const float*)d_in[2];  // 1*32*56*56
    float* out        = (float*)d_out;          // 32*32*56*56

    dim3 grid(Ww / TW, Hh);   // 14 x 56 = 784 workgroups
    lc2d_wmma_f32<<<grid, 128, 0, stream>>>(x, wgt, bias, out);
}